// SymmetricGaussianLoss_67834713473714
// MI455X (gfx1250) — compile-verified
//
#include <hip/hip_runtime.h>
#include <hip/hip_bf16.h>

// ---------------------------------------------------------------------------
// SymmetricGaussianLoss on MI455X (gfx1250, wave32, WMMA)
//
// maha_p[n,k] = f(n) . g(k)   (10-dim, padded to 16)
// maha_c[n,k] = u(n) . v(k)   (10-dim, padded to 16)
// Sum_k A[n,k]*maha_* is a GEMM  A[N,K] x [G|V][K,16] done with
// V_WMMA_F32_16X16X4_F32, then a per-row 16-dim dot + global reduction.
// ---------------------------------------------------------------------------

typedef __attribute__((ext_vector_type(2))) float v2f;
typedef __attribute__((ext_vector_type(8))) float v8f;

#define GL_EPS 1e-6f
#define KMAX 512          // LDS staging capacity for G|V (K <= 512)
#define WAVES_PER_WG 8    // 8 wave32 = 256 threads, 8 row-tiles per workgroup

// Analytic 3x3 inverse via adjugate/determinant (matches reference _inv3,
// with EPS added on the diagonal).
__device__ __forceinline__ void inv3_eps(const float* S, float* M) {
  float m00 = S[0] + GL_EPS, m01 = S[1],          m02 = S[2];
  float m10 = S[3],          m11 = S[4] + GL_EPS, m12 = S[5];
  float m20 = S[6],          m21 = S[7],          m22 = S[8] + GL_EPS;
  float c00 = m11*m22 - m12*m21;
  float c01 = m02*m21 - m01*m22;
  float c02 = m01*m12 - m02*m11;
  float c10 = m12*m20 - m10*m22;
  float c11 = m00*m22 - m02*m20;
  float c12 = m02*m10 - m00*m12;
  float c20 = m10*m21 - m11*m20;
  float c21 = m01*m20 - m00*m21;
  float c22 = m00*m11 - m01*m10;
  float det = m00*c00 + m01*c01 + m02*c02;
  float r = 1.0f / det;
  M[0]=c00*r; M[1]=c01*r; M[2]=c02*r;
  M[3]=c10*r; M[4]=c11*r; M[5]=c12*r;
  M[6]=c20*r; M[7]=c21*r; M[8]=c22*r;
}

__global__ void sgl_init_accum(float* accum) {
  if (threadIdx.x < 4) accum[threadIdx.x] = 0.0f;
}

// Parent-side features: g(k) (matrix features for maha_p) and
// v(k) (position features for maha_c).  One thread per (b,k).
__global__ void sgl_parent_feat(const float* __restrict__ mu_p,
                                const float* __restrict__ Sig_p,
                                float* __restrict__ G,
                                float* __restrict__ V,
                                int BK) {
  int i = blockIdx.x * blockDim.x + threadIdx.x;
  if (i >= BK) return;
  float p0 = mu_p[i*3+0], p1 = mu_p[i*3+1], p2 = mu_p[i*3+2];
  float M[9];
  inv3_eps(Sig_p + (size_t)i*9, M);
  // q = M p (M symmetric)
  float q0 = M[0]*p0 + M[1]*p1 + M[2]*p2;
  float q1 = M[3]*p0 + M[4]*p1 + M[5]*p2;
  float q2 = M[6]*p0 + M[7]*p1 + M[8]*p2;
  float pMp = p0*q0 + p1*q1 + p2*q2;
  float* g = G + (size_t)i*16;
  g[0]=M[0]; g[1]=M[4]; g[2]=M[8];
  g[3]=M[1]+M[3]; g[4]=M[2]+M[6]; g[5]=M[5]+M[7];   // 2*offdiag (sym)
  g[6]=-2.0f*q0; g[7]=-2.0f*q1; g[8]=-2.0f*q2;
  g[9]=pMp;
  g[10]=0.f; g[11]=0.f; g[12]=0.f; g[13]=0.f; g[14]=0.f; g[15]=0.f;
  float* v = V + (size_t)i*16;
  v[0]=p0*p0; v[1]=p1*p1; v[2]=p2*p2;
  v[3]=p0*p1; v[4]=p0*p2; v[5]=p1*p2;
  v[6]=p0; v[7]=p1; v[8]=p2; v[9]=1.0f;
  v[10]=0.f; v[11]=0.f; v[12]=0.f; v[13]=0.f; v[14]=0.f; v[15]=0.f;
}

// Child-side features: f(n) (position features, mask folded in) and
// u(n) (matrix features, mask folded in).  Also accumulates sum(mask).
__global__ void sgl_child_feat(const float* __restrict__ mu_c,
                               const float* __restrict__ Sig_c,
                               const float* __restrict__ mask,
                               float* __restrict__ F,
                               float* __restrict__ U,
                               float* __restrict__ accum,
                               int BN) {
  int i = blockIdx.x * blockDim.x + threadIdx.x;
  if (i >= BN) return;
  float c0 = mu_c[i*3+0], c1 = mu_c[i*3+1], c2 = mu_c[i*3+2];
  float mk = mask[i];
  float M[9];
  inv3_eps(Sig_c + (size_t)i*9, M);
  float r0 = M[0]*c0 + M[1]*c1 + M[2]*c2;
  float r1 = M[3]*c0 + M[4]*c1 + M[5]*c2;
  float r2 = M[6]*c0 + M[7]*c1 + M[8]*c2;
  float cMc = c0*r0 + c1*r1 + c2*r2;
  float* f = F + (size_t)i*16;
  f[0]=mk*c0*c0; f[1]=mk*c1*c1; f[2]=mk*c2*c2;
  f[3]=mk*c0*c1; f[4]=mk*c0*c2; f[5]=mk*c1*c2;
  f[6]=mk*c0; f[7]=mk*c1; f[8]=mk*c2; f[9]=mk;
  f[10]=0.f; f[11]=0.f; f[12]=0.f; f[13]=0.f; f[14]=0.f; f[15]=0.f;
  float* u = U + (size_t)i*16;
  u[0]=mk*M[0]; u[1]=mk*M[4]; u[2]=mk*M[8];
  u[3]=mk*(M[1]+M[3]); u[4]=mk*(M[2]+M[6]); u[5]=mk*(M[5]+M[7]);
  u[6]=-2.0f*mk*r0; u[7]=-2.0f*mk*r1; u[8]=-2.0f*mk*r2;
  u[9]=mk*cMc;
  u[10]=0.f; u[11]=0.f; u[12]=0.f; u[13]=0.f; u[14]=0.f; u[15]=0.f;
  atomicAdd(&accum[2], mk);
}

// Main pass: stream A once, contract over K with WMMA f32 16x16x4.
// One workgroup = 8 waves = 8 independent 16-row tiles of one batch b.
__global__ __launch_bounds__(256) void sgl_main(const float* __restrict__ A,
                                                const float* __restrict__ G,
                                                const float* __restrict__ V,
                                                const float* __restrict__ F,
                                                const float* __restrict__ U,
                                                float* __restrict__ accum,
                                                int N, int K) {
  __shared__ float sG[KMAX * 16];
  __shared__ float sV[KMAX * 16];

  const int tilesPerB = N >> 4;
  const int wgPerB    = tilesPerB / WAVES_PER_WG;
  const int b         = blockIdx.x / wgPerB;
  const int tg        = blockIdx.x % wgPerB;

  // Stage this batch's parent feature panels (K x 16 each) into LDS.
  const float4* Gb = (const float4*)(G + (size_t)b * K * 16);
  const float4* Vb = (const float4*)(V + (size_t)b * K * 16);
  const int nvec = K * 4;  // K*16 floats / 4
  for (int idx = threadIdx.x; idx < nvec; idx += blockDim.x) {
    ((float4*)sG)[idx] = Gb[idx];
    ((float4*)sV)[idx] = Vb[idx];
  }
  __syncthreads();

  const int wave  = threadIdx.x >> 5;
  const int lane  = threadIdx.x & 31;
  const int row   = lane & 15;     // A-fragment row / B-fragment column
  const int khalf = lane >> 4;     // 0: K pair {0,1}; 1: K pair {2,3}

  const int    n0  = (tg * WAVES_PER_WG + wave) << 4;  // tile row base in b
  const size_t bn0 = (size_t)b * N + n0;
  const float* Arow = A + (bn0 + row) * (size_t)K + 2 * khalf;

  v8f accP = {};  // 16 rows x 16 parent-matrix features
  v8f accV = {};  // 16 rows x 16 parent-position features

  for (int kc = 0; kc < K; kc += 4) {
    // A fragment: lane holds A[n0+row, kc+2*khalf + {0,1}]
    float2 a2 = *(const float2*)(Arow + kc);
    v2f av; av.x = a2.x; av.y = a2.y;
    __builtin_prefetch(Arow + kc + 64, 0, 0);  // global_prefetch ~16 iters ahead
    // B fragments from LDS (v0: K = kc+2*khalf, v1: K = kc+2*khalf+1)
    const int kb = kc + 2 * khalf;
    v2f bg; bg.x = sG[kb*16 + row]; bg.y = sG[(kb+1)*16 + row];
    v2f bv; bv.x = sV[kb*16 + row]; bv.y = sV[(kb+1)*16 + row];
    accP = __builtin_amdgcn_wmma_f32_16x16x4_f32(false, av, false, bg,
                                                 (short)0, accP, false, false);
    accV = __builtin_amdgcn_wmma_f32_16x16x4_f32(false, av, false, bv,
                                                 (short)0, accV, false, false);
  }

  // Epilogue: dist_p += f(n) . accP[n,:], dist_c += u(n) . accV[n,:]
  // C/D layout: lane t = lane&15 holds column t; VGPR i holds row i (+8 if lane>=16).
  const int t      = lane & 15;
  const int rowoff = khalf * 8;
  float pp = 0.0f, pc = 0.0f;
#pragma unroll
  for (int i = 0; i < 8; ++i) {
    const size_t m = bn0 + rowoff + i;
    pp += accP[i] * F[m * 16 + t];
    pc += accV[i] * U[m * 16 + t];
  }
  // wave32 butterfly reduction
#pragma unroll
  for (int off = 16; off > 0; off >>= 1) {
    pp += __shfl_xor(pp, off, 32);
    pc += __shfl_xor(pc, off, 32);
  }
  if (lane == 0) {
    atomicAdd(&accum[0], pp);
    atomicAdd(&accum[1], pc);
  }
}

__global__ void sgl_finalize(const float* __restrict__ accum,
                             float* __restrict__ out) {
  if (threadIdx.x == 0) {
    float denom = fmaxf(accum[2], 1.0f);
    float sp = accum[0] / denom;
    float sc = accum[1] / denom;
    out[0] = sp + sc;  // final_loss
    out[1] = sp;       // dist_p_m
    out[2] = sc;       // dist_c_m
  }
}

extern "C" void kernel_launch(void* const* d_in, const int* in_sizes, int n_in,
                              void* d_out, int out_size, void* d_ws, size_t ws_size,
                              hipStream_t stream) {
  const float* mu_c  = (const float*)d_in[0];  // [B,N,3]
  const float* Sig_c = (const float*)d_in[1];  // [B,N,3,3]
  const float* mu_p  = (const float*)d_in[2];  // [B,K,3]
  const float* Sig_p = (const float*)d_in[3];  // [B,K,3,3]
  const float* A     = (const float*)d_in[4];  // [B,N,K]
  const float* mask  = (const float*)d_in[5];  // [B,N]
  float* out = (float*)d_out;

  const long long BN  = in_sizes[5];
  const long long BNK = in_sizes[4];
  const long long BK  = in_sizes[2] / 3;
  const int K = (int)(BNK / BN);
  const int B = (int)(BK / K);
  const int N = (int)(BN / B);

  float* ws    = (float*)d_ws;
  float* accum = ws;                         // [4]: Sp, Sc, maskSum, pad
  float* G     = ws + 16;                    // [B*K,16]
  float* V     = G + (size_t)B * K * 16;     // [B*K,16]
  float* F     = V + (size_t)B * K * 16;     // [B*N,16]
  float* U     = F + (size_t)B * N * 16;     // [B*N,16]

  sgl_init_accum<<<1, 32, 0, stream>>>(accum);
  sgl_parent_feat<<<(int)((BK + 255) / 256), 256, 0, stream>>>(mu_p, Sig_p, G, V, (int)BK);
  sgl_child_feat<<<(int)((BN + 255) / 256), 256, 0, stream>>>(mu_c, Sig_c, mask, F, U, accum, (int)BN);
  const int wgPerB = (N / 16) / WAVES_PER_WG;
  sgl_main<<<B * wgPerB, 256, 0, stream>>>(A, G, V, F, U, accum, N, K);
  sgl_finalize<<<1, 32, 0, stream>>>(accum, out);
}